// LSTMnetwork_12008728559800
// MI455X (gfx1250) — compile-verified
//
#include <hip/hip_runtime.h>
#include <hip/hip_bf16.h>
#include <math.h>

// ---------------------------------------------------------------------------
// Sequential LSTM scan on one WGP (latency-bound problem):
//  - w_hh (400x100 f32) lives in VGPRs as WMMA A-tiles, 25 waves x 25 tiles.
//  - h @ w_hh^T via V_WMMA_F32_16X16X4_F32 with B columns replicated, so every
//    lane holds its half-wave's 8 rows of the 16-row result slice (no reduce).
//  - B fragment fetched with one half-wave-uniform ds_load_b64 (no cndmask).
//  - gates on threads 0..99: hardware v_tanh_f32, sigmoid = 0.5*tanh(x/2)+0.5.
//  - seq staged through LDS in 4KB chunks, prefetch for the next chunk.
// ---------------------------------------------------------------------------

typedef float v2f __attribute__((ext_vector_type(2)));
typedef float v4f __attribute__((ext_vector_type(4)));
typedef float v8f __attribute__((ext_vector_type(8)));

#define T_LEN   262144
#define HDIM    100
#define GDIM    400      // 4*H
#define KCH     25       // 100 / 4 K-chunks per WMMA row-tile
#define NWAVES  25       // 400 rows / 16 rows-per-wave
#define NTH     800      // 25 waves * 32 lanes
#define SCH     1024     // seq LDS chunk (elements); T_LEN % SCH == 0

__device__ __forceinline__ float fast_tanh(float x) {
#if __has_builtin(__builtin_amdgcn_tanhf)
    return __builtin_amdgcn_tanhf(x);        // v_tanh_f32 (gfx1250 trans op)
#else
    return tanhf(x);
#endif
}

__device__ __forceinline__ float fast_sigmoid(float x) {
    // sigmoid(x) = 0.5 * tanh(x/2) + 0.5  -> one trans op + one FMA
    return fmaf(fast_tanh(0.5f * x), 0.5f, 0.5f);
}

__global__ __launch_bounds__(NTH, 1)
void lstm_seq_wmma(const float* __restrict__ seq,
                   const float* __restrict__ w_ih,   // [400,1]
                   const float* __restrict__ w_hh,   // [400,100] row-major
                   const float* __restrict__ b_ih,   // [400]
                   const float* __restrict__ b_hh,   // [400]
                   const float* __restrict__ w_lin,  // [1,100]
                   const float* __restrict__ b_lin,  // [1]
                   float* __restrict__ out)          // [1]
{
    __shared__ __align__(16) float s_h[104];     // hidden state (padded)
    __shared__ __align__(16) float s_g[GDIM];    // pre-activation gates
    __shared__ __align__(16) float s_seq[SCH];   // staged input chunk

    const int tid  = (int)threadIdx.x;
    const int wave = tid >> 5;       // gfx1250 is wave32
    const int lane = tid & 31;

    // ---- load W_hh into registers as WMMA A-tiles (16x4 f32 layout) --------
    // A 16x4 f32: VGPR0 = lanes 0-15 (M=lane,K=0), lanes 16-31 (M=lane-16,K=2)
    //             VGPR1 = K=1 / K=3 respectively.
    const int mrow = (lane < 16) ? lane : (lane - 16);
    const int koff = (lane < 16) ? 0 : 2;
    const int row  = wave * 16 + mrow;          // 0..399

    v2f wa[KCH];
#pragma unroll
    for (int kc = 0; kc < KCH; ++kc) {
        const float* p = w_hh + row * HDIM + 4 * kc + koff;
        wa[kc].x = p[0];
        wa[kc].y = p[1];
    }

    // ---- per-lane loop-invariant input-projection coefficients -------------
    // Lanes 0 and 16 of wave w scatter rows [16w..16w+7] / [16w+8..16w+15].
    const int rbase = wave * 16 + ((lane >= 16) ? 8 : 0);
    float wihr[8], bsr[8];
#pragma unroll
    for (int j = 0; j < 8; ++j) {
        wihr[j] = w_ih[rbase + j];
        bsr[j]  = b_ih[rbase + j] + b_hh[rbase + j];
    }

    // ---- init LDS ----------------------------------------------------------
    if (tid < 104) s_h[tid] = 0.0f;

    // Per-lane B base pointer: lanes 0-15 read h[4k..4k+1], 16-31 h[4k+2..+3].
    const float* hb = &s_h[koff];

    float cstate = 0.0f;             // cell state, threads 0..99 only
    __syncthreads();

    // ---- sequential scan ---------------------------------------------------
    for (int t = 0; t < T_LEN; ++t) {
        const int tc = t & (SCH - 1);
        if (tc == 0) {
            // stage this chunk of seq into LDS; prefetch the next chunk
            for (int i = tid; i < SCH; i += NTH)
                s_seq[i] = seq[t + i];
            if (t + SCH < T_LEN)
                __builtin_prefetch(&seq[t + SCH + tid], 0, 3);
            __syncthreads();
        }

        // ---- g[16w .. 16w+15] = W_tile @ h  (25 chained f32 WMMAs) ---------
        // B (4x16) replicated across N: every column computes the same dot,
        // so lanes 0-15 hold the wave's results 0-7 and lanes 16-31 hold 8-15.
        v8f acc = {};
#pragma unroll
        for (int kc = 0; kc < KCH; ++kc) {
            v2f b = *(const v2f*)(hb + 4 * kc);   // one ds_load_b64, no select
            acc = __builtin_amdgcn_wmma_f32_16x16x4_f32(
                      false, wa[kc], false, b, (short)0, acc, false, false);
        }

        // ---- scatter g (+ input projection + biases) to LDS ----------------
        const float x = s_seq[tc];
        if (lane == 0 || lane == 16) {
            v4f g0, g1;
#pragma unroll
            for (int j = 0; j < 4; ++j) {
                g0[j] = fmaf(x, wihr[j],     acc[j]     + bsr[j]);
                g1[j] = fmaf(x, wihr[4 + j], acc[4 + j] + bsr[4 + j]);
            }
            *(v4f*)&s_g[rbase]     = g0;          // two ds_store_b128
            *(v4f*)&s_g[rbase + 4] = g1;
        }
        __syncthreads();

        // ---- gate nonlinearities + state update (threads 0..99) ------------
        if (tid < HDIM) {
            const float i_ = fast_sigmoid(s_g[tid]);
            const float f_ = fast_sigmoid(s_g[tid + HDIM]);
            const float g_ = fast_tanh(s_g[tid + 2 * HDIM]);
            const float o_ = fast_sigmoid(s_g[tid + 3 * HDIM]);
            cstate = fmaf(f_, cstate, i_ * g_);
            s_h[tid] = o_ * fast_tanh(cstate);
        }
        __syncthreads();
    }

    // ---- final linear projection: out = h_last . w_lin + b_lin -------------
    if (tid < HDIM) s_g[tid] = s_h[tid] * w_lin[tid];
    __syncthreads();
    if (tid == 0) {
        float acc = b_lin[0];
        for (int u = 0; u < HDIM; ++u) acc += s_g[u];
        out[0] = acc;
    }
}

extern "C" void kernel_launch(void* const* d_in, const int* in_sizes, int n_in,
                              void* d_out, int out_size, void* d_ws, size_t ws_size,
                              hipStream_t stream) {
    (void)in_sizes; (void)n_in; (void)out_size; (void)d_ws; (void)ws_size;
    const float* seq   = (const float*)d_in[0];
    const float* w_ih  = (const float*)d_in[1];
    const float* w_hh  = (const float*)d_in[2];
    const float* b_ih  = (const float*)d_in[3];
    const float* b_hh  = (const float*)d_in[4];
    const float* w_lin = (const float*)d_in[5];
    const float* b_lin = (const float*)d_in[6];
    float* out = (float*)d_out;

    lstm_seq_wmma<<<1, NTH, 0, stream>>>(seq, w_ih, w_hh, b_ih, b_hh,
                                         w_lin, b_lin, out);
}